// GATModel_72885595013673
// MI455X (gfx1250) — compile-verified
//
#include <hip/hip_runtime.h>
#include <hip/hip_bf16.h>
#include <math.h>

// ---------------- problem constants (fixed by reference) ----------------
#define HEADS   8
#define HID     16
#define DFEAT   128          // HEADS*HID
#define NPG     32           // nodes per graph
#define DEG     8
#define EPG     (NPG*DEG)    // 256 edges per graph
#define NGRAPH  2048
#define NNODES  (NGRAPH*NPG) // 65536
#define NEDGES  (NNODES*DEG) // 524288
#define INDIM   7
#define AG      5
#define OUTD    2
#define SLOPE   0.2f
#define EPS_LN  1e-5f

typedef __attribute__((ext_vector_type(2))) float v2f;
typedef __attribute__((ext_vector_type(8))) float v8f;
typedef __attribute__((ext_vector_type(4))) int   v4i;

// ---------------------------------------------------------------------
// Async global->LDS DMA (GLOBAL_LOAD_ASYNC_TO_LDS_B128, ASYNCcnt) with
// compile-time fallback to VGPR-mediated float4 staging if this
// toolchain does not declare the builtins. Probe-confirmed signature:
//   void __builtin_amdgcn_global_load_async_to_lds_b128(
//       v4i addrspace(1)* src, v4i addrspace(3)* dst, Ii offset, Ii cpol)
// ---------------------------------------------------------------------
#if __has_builtin(__builtin_amdgcn_global_load_async_to_lds_b128) && \
    __has_builtin(__builtin_amdgcn_s_wait_asynccnt)
#define USE_ASYNC_LDS 1
#else
#define USE_ASYNC_LDS 0
#endif

// Copy cnt16 float4-chunks from global gsrc into LDS ldst, cooperatively
// over 256 threads. Caller must __syncthreads() after (helper performs the
// per-thread async wait).
__device__ __forceinline__ void stage_tile_to_lds(float* ldst, const float* gsrc,
                                                  int cnt16, int t) {
#if USE_ASYNC_LDS
  for (int i = t; i < cnt16; i += 256) {
    __builtin_amdgcn_global_load_async_to_lds_b128(
        (__attribute__((address_space(1))) v4i*)(uintptr_t)(gsrc + i * 4),
        (__attribute__((address_space(3))) v4i*)(ldst + i * 4),
        /*imm offset*/0, /*cpol*/0);
  }
  __builtin_amdgcn_s_wait_asynccnt(0);
#else
  const float4* s4 = (const float4*)gsrc;
  float4* d4 = (float4*)ldst;
  for (int i = t; i < cnt16; i += 256) d4[i] = s4[i];
#endif
}

// =======================================================================
// Kernel 1: conv1 input projection  xl = x(N x 7) @ W(7 x 128)
// K=7 -> pure VALU, memory bound.
// =======================================================================
__global__ __launch_bounds__(256) void in_gemm_kernel(
    const float* __restrict__ x, const float* __restrict__ W,
    float* __restrict__ xl) {
  int i = blockIdx.x * 256 + threadIdx.x;       // over NNODES*128
  int n = i >> 7, c = i & 127;
  const float* xr = x + n * INDIM;
  float acc = 0.f;
#pragma unroll
  for (int k = 0; k < INDIM; ++k) acc += xr[k] * W[k * DFEAT + c];
  xl[i] = acc;
}

// =======================================================================
// Kernel 2: xl = H(N x 128) @ W(128 x 128) using V_WMMA_F32_16X16X4_F32.
// Block = 256 threads = 8 waves. W is staged transposed in LDS (64 KB)
// so each B fragment is one contiguous ds b64 read. Each wave owns one
// 16-row M-tile and all 8 N-tiles (A fragment reused 8x per K-step).
//
// Fragment layouts (CDNA5 ISA 7.12.2, wave32):
//   A 16x4 f32 : lane L, vgpr r -> m = L%16, k = r + 2*(L/16)
//   B  4x16 f32: lane L, vgpr r -> n = L%16, k = r + 2*(L/16)
//   C 16x16 f32: lane L, vgpr r -> m = r + 8*(L/16), n = L%16
// =======================================================================
__global__ __launch_bounds__(256) void gemm128_wmma_kernel(
    const float* __restrict__ A, const float* __restrict__ W,
    float* __restrict__ C) {
  __shared__ __align__(16) float Ws[DFEAT * DFEAT];   // transposed: Ws[n*128+k]
  const int t = threadIdx.x;
  for (int i = t; i < DFEAT * DFEAT; i += 256) {
    int k = i >> 7, n = i & 127;
    Ws[n * DFEAT + k] = W[i];
  }
  __syncthreads();

  const int wave  = t >> 5;
  const int lane  = t & 31;
  const int mtile = blockIdx.x * 8 + wave;      // 16-row tile
  const int mrow  = mtile * 16 + (lane & 15);
  const int khalf = (lane >> 4) * 2;            // 0 or 2
  const int nlane = lane & 15;
  const float* arow = A + (size_t)mrow * DFEAT;

  v8f acc[8];
#pragma unroll
  for (int j = 0; j < 8; ++j) acc[j] = (v8f){0.f,0.f,0.f,0.f,0.f,0.f,0.f,0.f};

  for (int k = 0; k < DFEAT; k += 4) {
    v2f a = *(const v2f*)(arow + k + khalf);     // global b64, coalesced
#pragma unroll
    for (int j = 0; j < 8; ++j) {
      v2f b = *(const v2f*)(&Ws[(j * 16 + nlane) * DFEAT + k + khalf]);  // ds b64
      acc[j] = __builtin_amdgcn_wmma_f32_16x16x4_f32(
          false, a, false, b, (short)0, acc[j], false, false);
    }
  }

  const int mbase = mtile * 16 + 8 * (lane >> 4);
#pragma unroll
  for (int j = 0; j < 8; ++j)
#pragma unroll
    for (int r = 0; r < 8; ++r)
      C[(size_t)(mbase + r) * DFEAT + j * 16 + nlane] = acc[j][r];
}

// =======================================================================
// Kernel 3: fused per-graph GAT layer. One block per graph (32 nodes,
// 256 edges [+32 self loops]). Everything in LDS: attention logits,
// edge softmax (LDS atomicMax with order-preserving int encoding,
// LDS ds_add_f32 sums), scatter-aggregation, bias + LayerNorm + ReLU.
// =======================================================================
__device__ __forceinline__ int enc_f32(float f) {
  int b = __float_as_int(f);
  return b >= 0 ? b : (b ^ 0x7FFFFFFF);          // monotone wrt float order
}
__device__ __forceinline__ float dec_f32(int e) {
  return __int_as_float(e >= 0 ? e : (e ^ 0x7FFFFFFF));
}

__global__ __launch_bounds__(256) void gat_graph_kernel(
    const float* __restrict__ xl,       // [N,128] projected features
    const float* __restrict__ eattr,    // [E]
    const int*   __restrict__ esrc,     // [E]
    const int*   __restrict__ edst,     // [E]
    const float* __restrict__ att_src,  // [8,16]
    const float* __restrict__ att_dst,  // [8,16]
    const float* __restrict__ att_e,    // [8,16]
    const float* __restrict__ W_e,      // [1,128]
    const float* __restrict__ bias,     // [128]
    const float* __restrict__ ln_g,     // [128]
    const float* __restrict__ ln_b,     // [128]
    float* __restrict__ hout,           // [N,128]
    int self_loops) {
  __shared__ __align__(16) float sxl [NPG * DFEAT];
  __shared__ float sout[NPG * DFEAT];
  __shared__ float salpha[(EPG + NPG) * HEADS];
  __shared__ float sasrc[NPG * HEADS], sadst[NPG * HEADS];
  __shared__ int   smax [NPG * HEADS];
  __shared__ float ssum [NPG * HEADS];
  __shared__ int   ssrc [EPG + NPG], sdst[EPG + NPG];
  __shared__ float sea  [EPG + NPG];
  __shared__ float swedot[HEADS];
  __shared__ float sattr[NPG], scnt[NPG];
  __shared__ float smu[NPG], srs[NPG];

  const int g = blockIdx.x, t = threadIdx.x;
  const int nodeBase = g * NPG;
  const int edgeBase = g * EPG;
  const int nTot = self_loops ? (EPG + NPG) : EPG;

  // stage the 32x128 feature tile via async DMA (no VGPR round-trip)
  stage_tile_to_lds(sxl, xl + (size_t)nodeBase * DFEAT, NPG * DFEAT / 4, t);

  if (t < EPG) {
    ssrc[t] = esrc[edgeBase + t] - nodeBase;
    sdst[t] = edst[edgeBase + t] - nodeBase;
    sea [t] = eattr[edgeBase + t];
  }
  if (t < HEADS) {           // a_e[e,h] = eattr[e] * dot(W_e[h], att_e[h])
    float acc = 0.f;
    for (int c = 0; c < HID; ++c) acc += W_e[t * HID + c] * att_e[t * HID + c];
    swedot[t] = acc;
  }
  for (int i = t; i < NPG * DFEAT; i += 256) sout[i] = 0.f;
  if (t < NPG * HEADS) { smax[t] = 0x807FFFFF /* enc(-inf) */; ssum[t] = 0.f; }
  if (t < NPG) { sattr[t] = 0.f; scnt[t] = 0.f; }
  __syncthreads();

  if (self_loops) {          // fill_value='mean' self-loop edge attrs
    if (t < EPG) {
      atomicAdd(&sattr[sdst[t]], sea[t]);     // ds_add_f32
      atomicAdd(&scnt [sdst[t]], 1.0f);
    }
    __syncthreads();
    if (t < NPG) {
      ssrc[EPG + t] = t; sdst[EPG + t] = t;
      sea [EPG + t] = sattr[t] / fmaxf(scnt[t], 1.0f);
    }
  }
  if (t < NPG * HEADS) {     // per-node logit halves
    int n = t >> 3, h = t & 7;
    const float* xr = &sxl[n * DFEAT + h * HID];
    float as_ = 0.f, ad_ = 0.f;
#pragma unroll
    for (int c = 0; c < HID; ++c) {
      as_ += xr[c] * att_src[h * HID + c];
      ad_ += xr[c] * att_dst[h * HID + c];
    }
    sasrc[t] = as_; sadst[t] = ad_;
  }
  __syncthreads();

  // pass 1: leaky-relu logits + segment max over dst
  for (int e = t; e < nTot; e += 256) {
    int s = ssrc[e], d = sdst[e];
    float eav = sea[e];
#pragma unroll
    for (int h = 0; h < HEADS; ++h) {
      float a = sasrc[s * HEADS + h] + sadst[d * HEADS + h] + eav * swedot[h];
      a = a > 0.f ? a : SLOPE * a;
      salpha[e * HEADS + h] = a;
      atomicMax(&smax[d * HEADS + h], enc_f32(a));   // ds_max_i32
    }
  }
  __syncthreads();

  // pass 2: exp(alpha - max) + segment sum
  for (int e = t; e < nTot; e += 256) {
    int d = sdst[e];
#pragma unroll
    for (int h = 0; h < HEADS; ++h) {
      float m  = dec_f32(smax[d * HEADS + h]);
      float ex = __expf(salpha[e * HEADS + h] - m);
      salpha[e * HEADS + h] = ex;
      atomicAdd(&ssum[d * HEADS + h], ex);
    }
  }
  __syncthreads();

  // pass 3: att-weighted scatter of source features
  for (int e = t; e < nTot; e += 256) {
    int s = ssrc[e], d = sdst[e];
    float att[HEADS];
#pragma unroll
    for (int h = 0; h < HEADS; ++h)
      att[h] = salpha[e * HEADS + h] / (ssum[d * HEADS + h] + 1e-16f);
    for (int c = 0; c < DFEAT; ++c)
      atomicAdd(&sout[d * DFEAT + c], att[c >> 4] * sxl[s * DFEAT + c]);
  }
  __syncthreads();

  // bias + LayerNorm stats
  if (t < NPG) {
    float sum = 0.f, sq = 0.f;
    for (int c = 0; c < DFEAT; ++c) {
      float v = sout[t * DFEAT + c] + bias[c];
      sum += v; sq += v * v;
    }
    float mu  = sum * (1.0f / DFEAT);
    float var = sq * (1.0f / DFEAT) - mu * mu;
    smu[t] = mu; srs[t] = rsqrtf(var + EPS_LN);
  }
  __syncthreads();
  for (int i = t; i < NPG * DFEAT; i += 256) {
    int n = i >> 7, c = i & 127;
    float v = (sout[i] + bias[c] - smu[n]) * srs[n] * ln_g[c] + ln_b[c];
    hout[(size_t)nodeBase * DFEAT + i] = fmaxf(v, 0.f);
  }
}

// =======================================================================
// Kernel 4: per-graph head. mean-pool, [agent|graph] concat, FC1+ReLU,
// LayerNorm, FC2. Tiny per graph -> VALU in LDS.
// =======================================================================
__global__ __launch_bounds__(256) void head_kernel(
    const float* __restrict__ h,     // [N,128]
    const float* __restrict__ fc1W,  // [256,64]
    const float* __restrict__ fc1b,  // [64]
    const float* __restrict__ lng,   // [64]
    const float* __restrict__ lnb,   // [64]
    const float* __restrict__ fc2W,  // [64,2]
    const float* __restrict__ fc2b,  // [2]
    float* __restrict__ out) {       // [G,5,2]
  __shared__ __align__(16) float sh[NPG * DFEAT];
  __shared__ float sg[DFEAT];
  __shared__ float sf[AG * 64];
  __shared__ float smu[AG], srs[AG];
  const int g = blockIdx.x, t = threadIdx.x;

  stage_tile_to_lds(sh, h + (size_t)g * NPG * DFEAT, NPG * DFEAT / 4, t);
  __syncthreads();

  if (t < DFEAT) {                    // global mean pool
    float s = 0.f;
    for (int n = 0; n < NPG; ++n) s += sh[n * DFEAT + t];
    sg[t] = s * (1.0f / NPG);
  }
  __syncthreads();

  for (int oi = t; oi < AG * 64; oi += 256) {   // FC1 + ReLU
    int a = oi >> 6, j = oi & 63;
    float acc = fc1b[j];
    for (int k = 0; k < DFEAT; ++k) acc += sh[a * DFEAT + k] * fc1W[k * 64 + j];
    for (int k = 0; k < DFEAT; ++k) acc += sg[k] * fc1W[(DFEAT + k) * 64 + j];
    sf[oi] = fmaxf(acc, 0.f);
  }
  __syncthreads();
  if (t < AG) {                       // LN stats over 64
    float sum = 0.f, sq = 0.f;
    for (int j = 0; j < 64; ++j) { float v = sf[t * 64 + j]; sum += v; sq += v * v; }
    float mu = sum * (1.f / 64.f), var = sq * (1.f / 64.f) - mu * mu;
    smu[t] = mu; srs[t] = rsqrtf(var + EPS_LN);
  }
  __syncthreads();
  for (int oi = t; oi < AG * 64; oi += 256) {
    int a = oi >> 6, j = oi & 63;
    sf[oi] = (sf[oi] - smu[a]) * srs[a] * lng[j] + lnb[j];
  }
  __syncthreads();
  if (t < AG * OUTD) {                // FC2
    int a = t >> 1, o = t & 1;
    float acc = fc2b[o];
    for (int j = 0; j < 64; ++j) acc += sf[a * 64 + j] * fc2W[j * 2 + o];
    out[(g * AG + a) * OUTD + o] = acc;
  }
}

// =======================================================================
// Launcher.
// Input order = insertion-order recursive flatten of setup_inputs():
//  0 x, 1 edge_attr,
//  params: conv{1,2,3}: {W, att_src, att_dst, W_e, att_e, bias} -> 2..19
//          ln{1,2,3}: {g,b} -> 20..25
//          26 fc1_W, 27 fc1_b, 28 fc1_ln.g, 29 fc1_ln.b, 30 fc2_W, 31 fc2_b
//  32 edge_index (2 x E), 33 batch, 34 num_graphs
// Workspace: 2 x N*128 f32 = 64 MB.
// =======================================================================
extern "C" void kernel_launch(void* const* d_in, const int* in_sizes, int n_in,
                              void* d_out, int out_size, void* d_ws, size_t ws_size,
                              hipStream_t stream) {
  (void)in_sizes; (void)n_in; (void)out_size; (void)ws_size;
  const float* x     = (const float*)d_in[0];
  const float* eattr = (const float*)d_in[1];
  const float* cv[3][6];
  for (int l = 0; l < 3; ++l)
    for (int p = 0; p < 6; ++p) cv[l][p] = (const float*)d_in[2 + l * 6 + p];
  const float* lng[3] = {(const float*)d_in[20], (const float*)d_in[22], (const float*)d_in[24]};
  const float* lnb[3] = {(const float*)d_in[21], (const float*)d_in[23], (const float*)d_in[25]};
  const float* fc1W = (const float*)d_in[26];
  const float* fc1b = (const float*)d_in[27];
  const float* flg  = (const float*)d_in[28];
  const float* flb  = (const float*)d_in[29];
  const float* fc2W = (const float*)d_in[30];
  const float* fc2b = (const float*)d_in[31];
  const int* eidx = (const int*)d_in[32];
  const int* esrc = eidx;
  const int* edst = eidx + NEDGES;
  float* out = (float*)d_out;

  float* bufXL = (float*)d_ws;                         // N*128
  float* bufH  = bufXL + (size_t)NNODES * DFEAT;       // N*128

  // indices into cv[l][*]: 0=W 1=att_src 2=att_dst 3=W_e 4=att_e 5=bias
  // ---- layer 1 (no self loops) ----
  in_gemm_kernel<<<NNODES * DFEAT / 256, 256, 0, stream>>>(x, cv[0][0], bufXL);
  gat_graph_kernel<<<NGRAPH, 256, 0, stream>>>(bufXL, eattr, esrc, edst,
      cv[0][1], cv[0][2], cv[0][4], cv[0][3], cv[0][5], lng[0], lnb[0], bufH, 0);
  // ---- layer 2 ----
  gemm128_wmma_kernel<<<NNODES / 128, 256, 0, stream>>>(bufH, cv[1][0], bufXL);
  gat_graph_kernel<<<NGRAPH, 256, 0, stream>>>(bufXL, eattr, esrc, edst,
      cv[1][1], cv[1][2], cv[1][4], cv[1][3], cv[1][5], lng[1], lnb[1], bufH, 1);
  // ---- layer 3 ----
  gemm128_wmma_kernel<<<NNODES / 128, 256, 0, stream>>>(bufH, cv[2][0], bufXL);
  gat_graph_kernel<<<NGRAPH, 256, 0, stream>>>(bufXL, eattr, esrc, edst,
      cv[2][1], cv[2][2], cv[2][4], cv[2][3], cv[2][5], lng[2], lnb[2], bufH, 1);
  // ---- head ----
  head_kernel<<<NGRAPH, 256, 0, stream>>>(bufH, fc1W, fc1b, flg, flb, fc2W, fc2b, out);
}